// HyperGCN_4088808865995
// MI455X (gfx1250) — compile-verified
//
#include <hip/hip_runtime.h>
#include <cmath>
#include <stdint.h>

typedef float v2f __attribute__((ext_vector_type(2)));
typedef float v8f __attribute__((ext_vector_type(8)));

#define N_NODES 200000
#define NNZ_E   (N_NODES * 32)
#define D_IN    128
#define D_HID   16
#define N_CLS   40
#define N_PAD   48
#define M_TILES (N_NODES / 16)   // 12500, exact
#define ROWSTR  132              // padded LDS row stride (floats): 528B, 16B-aligned, bank-conflict-free

// ---------------------------------------------------------------------------
// GEMM1: X1[200000,16] = H[200000,128] @ W1[128,16]
// 4 waves/block, one 16-row tile per wave. Each wave stages its 8KB A-tile
// into LDS with global_load_async_to_lds_b128 (coalesced 512B bursts,
// ASYNCcnt-tracked), then runs 32 steps of v_wmma_f32_16x16x4_f32 with
// A-fragments from LDS (ds_load_b64) and B-fragments from cached W1.
// ---------------------------------------------------------------------------
__global__ __launch_bounds__(128) void gemm1_wmma(const float* __restrict__ H,
                                                  const float* __restrict__ W1,
                                                  float* __restrict__ X1) {
    __shared__ float sh[4 * 16 * ROWSTR];             // 33792 B
    const int wslot = threadIdx.x >> 5;
    const int wave  = blockIdx.x * 4 + wslot;         // 3125*4 == 12500, no guard
    const int lane  = threadIdx.x & 31;
    const int half  = lane >> 4;                      // 0: K+0/K+1, 1: K+2/K+3
    const int l16   = lane & 15;

    // ---- async stage: 16 rows x 512B, one b128 per lane per row ----
    const float* shbase = &sh[wslot * 16 * ROWSTR];
    const uint32_t lds0 = (uint32_t)(uintptr_t)shbase + (uint32_t)(lane * 16);
    const char* g0 = (const char*)(H + (size_t)wave * 16 * D_IN) + lane * 16;
#pragma unroll
    for (int r = 0; r < 16; ++r) {
        const uint32_t lds = lds0 + r * (ROWSTR * 4);
        const char*    g   = g0 + r * (D_IN * 4);
        asm volatile("global_load_async_to_lds_b128 %0, %1, off"
                     :: "v"(lds), "v"(g) : "memory");
    }
    asm volatile("s_wait_asynccnt 0x0" ::: "memory");

    // ---- WMMA main loop ----
    const float* arow = shbase + l16 * ROWSTR;        // A row for this lane
    v8f c = {0.f, 0.f, 0.f, 0.f, 0.f, 0.f, 0.f, 0.f};
#pragma unroll
    for (int k = 0; k < D_IN; k += 4) {
        const int kk = k + half * 2;
        v2f a, b;
        a.x = arow[kk];                               // ds_load_b64, conflict-free
        a.y = arow[kk + 1];
        b.x = W1[kk * D_HID + l16];
        b.y = W1[(kk + 1) * D_HID + l16];
        c = __builtin_amdgcn_wmma_f32_16x16x4_f32(false, a, false, b,
                                                  (short)0, c, false, false);
    }
#pragma unroll
    for (int v = 0; v < 8; ++v) {
        const int r = wave * 16 + v + half * 8;       // C/D layout: VGPR v -> M=v / M=v+8
        X1[(size_t)r * D_HID + l16] = c[v];
    }
}

// ---------------------------------------------------------------------------
// SpMM layer 1: acc[rows[e], :] += vals[e] * X1[cols[e], :]   (d=16)
// 16 lanes per edge; X1/acc are L2-resident (12.8 MB each).
// ---------------------------------------------------------------------------
__global__ __launch_bounds__(256) void spmm16(const int* __restrict__ rows,
                                              const int* __restrict__ cols,
                                              const float* __restrict__ vals,
                                              const float* __restrict__ X,
                                              float* __restrict__ acc) {
    const int e    = blockIdx.x * 16 + (threadIdx.x >> 4);
    const int lane = threadIdx.x & 15;
    if (e >= NNZ_E) return;
    const int   r = rows[e];
    const int   c = cols[e];
    const float v = vals[e];
    atomicAdd(&acc[(size_t)r * D_HID + lane], v * X[(size_t)c * D_HID + lane]);
}

// H1 = relu(acc + b1), in place
__global__ __launch_bounds__(256) void bias_relu16(float* __restrict__ acc,
                                                   const float* __restrict__ b1) {
    const int i = blockIdx.x * 256 + threadIdx.x;
    if (i >= N_NODES * D_HID) return;
    acc[i] = fmaxf(acc[i] + b1[i & 15], 0.0f);
}

// Zero-pad W2[16,40] -> W2p[16,48] so GEMM2 needs no predication around WMMA.
__global__ __launch_bounds__(256) void pad_w2(const float* __restrict__ W2,
                                              float* __restrict__ W2p) {
    for (int i = threadIdx.x; i < D_HID * N_PAD; i += 256) {
        const int k = i / N_PAD, n = i % N_PAD;
        W2p[i] = (n < N_CLS) ? W2[k * N_CLS + n] : 0.0f;
    }
}

// ---------------------------------------------------------------------------
// GEMM2: X2[200000,48] = H1[200000,16] @ W2p[16,48]; 3 N-tiles, K=16.
// H1 is L2-resident; direct loads are fine here.
// ---------------------------------------------------------------------------
__global__ __launch_bounds__(256) void gemm2_wmma(const float* __restrict__ H1,
                                                  const float* __restrict__ W2p,
                                                  float* __restrict__ X2) {
    const int wave = blockIdx.x * 8 + (threadIdx.x >> 5);
    if (wave >= M_TILES) return;                      // wave-uniform guard
    const int lane = threadIdx.x & 31;
    const int half = lane >> 4;
    const int l16  = lane & 15;
    const float* arow = H1 + (size_t)(wave * 16 + l16) * D_HID;

    v8f c0 = {0.f,0.f,0.f,0.f,0.f,0.f,0.f,0.f};
    v8f c1 = c0, c2 = c0;
#pragma unroll
    for (int k = 0; k < D_HID; k += 4) {
        const int kk = k + half * 2;
        v2f a;
        a.x = arow[kk];
        a.y = arow[kk + 1];
        v2f b0, b1, b2;
        b0.x = W2p[kk * N_PAD + l16];        b0.y = W2p[(kk + 1) * N_PAD + l16];
        b1.x = W2p[kk * N_PAD + 16 + l16];   b1.y = W2p[(kk + 1) * N_PAD + 16 + l16];
        b2.x = W2p[kk * N_PAD + 32 + l16];   b2.y = W2p[(kk + 1) * N_PAD + 32 + l16];
        c0 = __builtin_amdgcn_wmma_f32_16x16x4_f32(false, a, false, b0, (short)0, c0, false, false);
        c1 = __builtin_amdgcn_wmma_f32_16x16x4_f32(false, a, false, b1, (short)0, c1, false, false);
        c2 = __builtin_amdgcn_wmma_f32_16x16x4_f32(false, a, false, b2, (short)0, c2, false, false);
    }
#pragma unroll
    for (int v = 0; v < 8; ++v) {
        const size_t r = (size_t)(wave * 16 + v + half * 8) * N_PAD;
        X2[r + l16]      = c0[v];
        X2[r + 16 + l16] = c1[v];
        X2[r + 32 + l16] = c2[v];
    }
}

// ---------------------------------------------------------------------------
// SpMM layer 2: out[rows[e], :40] += vals[e] * X2[cols[e], :40] (stride 48)
// 8 lanes per edge, 5 features each.
// ---------------------------------------------------------------------------
__global__ __launch_bounds__(256) void spmm40(const int* __restrict__ rows,
                                              const int* __restrict__ cols,
                                              const float* __restrict__ vals,
                                              const float* __restrict__ X2,
                                              float* __restrict__ out) {
    const int e    = blockIdx.x * 32 + (threadIdx.x >> 3);
    const int lane = threadIdx.x & 7;
    if (e >= NNZ_E) return;
    const int   r = rows[e];
    const int   c = cols[e];
    const float v = vals[e];
    const float* src = X2 + (size_t)c * N_PAD;
    float* dst = out + (size_t)r * N_CLS;
#pragma unroll
    for (int j = 0; j < 5; ++j) {
        const int f = lane + j * 8;
        atomicAdd(&dst[f], v * src[f]);
    }
}

// out = log_softmax(relu(out + b2)) per row, in place
__global__ __launch_bounds__(256) void bias_relu_logsoftmax(float* __restrict__ out,
                                                            const float* __restrict__ b2) {
    const int row = blockIdx.x * 256 + threadIdx.x;
    if (row >= N_NODES) return;
    float* p = out + (size_t)row * N_CLS;
    float y[N_CLS];
    float m = -3.402823e38f;
#pragma unroll
    for (int i = 0; i < N_CLS; ++i) {
        y[i] = fmaxf(p[i] + b2[i], 0.0f);
        m = fmaxf(m, y[i]);
    }
    float s = 0.0f;
#pragma unroll
    for (int i = 0; i < N_CLS; ++i) s += __expf(y[i] - m);
    const float lse = m + __logf(s);
#pragma unroll
    for (int i = 0; i < N_CLS; ++i) p[i] = y[i] - lse;
}

// ---------------------------------------------------------------------------
extern "C" void kernel_launch(void* const* d_in, const int* in_sizes, int n_in,
                              void* d_out, int out_size, void* d_ws, size_t ws_size,
                              hipStream_t stream) {
    const float* H    = (const float*)d_in[0];
    const int*   rows = (const int*)  d_in[1];
    const int*   cols = (const int*)  d_in[2];
    const float* vals = (const float*)d_in[3];
    const float* W1   = (const float*)d_in[4];
    const float* b1   = (const float*)d_in[5];
    const float* W2   = (const float*)d_in[6];
    const float* b2   = (const float*)d_in[7];
    float* out = (float*)d_out;

    // workspace layout: [X1 | aliased X2 padded] [H1 acc] [W2 padded]
    const size_t X2_BYTES = (size_t)N_NODES * N_PAD * sizeof(float);   // 38.4 MB
    const size_t H1_BYTES = (size_t)N_NODES * D_HID * sizeof(float);   // 12.8 MB
    float* Xbuf = (float*)d_ws;                                  // X1 then X2
    float* H1   = (float*)((char*)d_ws + X2_BYTES);
    float* W2p  = (float*)((char*)d_ws + X2_BYTES + H1_BYTES);

    hipMemsetAsync(H1, 0, H1_BYTES, stream);
    hipMemsetAsync(out, 0, (size_t)N_NODES * N_CLS * sizeof(float), stream);

    // layer 1
    gemm1_wmma<<<M_TILES / 4, 128, 0, stream>>>(H, W1, Xbuf);
    spmm16<<<NNZ_E / 16, 256, 0, stream>>>(rows, cols, vals, Xbuf, H1);
    bias_relu16<<<(N_NODES * D_HID + 255) / 256, 256, 0, stream>>>(H1, b1);

    // layer 2
    pad_w2<<<1, 256, 0, stream>>>(W2, W2p);
    gemm2_wmma<<<(M_TILES + 7) / 8, 256, 0, stream>>>(H1, W2p, Xbuf);
    spmm40<<<NNZ_E / 32, 256, 0, stream>>>(rows, cols, vals, Xbuf, out);

    // epilogue
    bias_relu_logsoftmax<<<(N_NODES + 255) / 256, 256, 0, stream>>>(out, b2);
}